// MPC_59339268161679
// MI455X (gfx1250) — compile-verified
//
#include <hip/hip_runtime.h>

typedef float v2f __attribute__((ext_vector_type(2)));
typedef float v8f __attribute__((ext_vector_type(8)));

constexpr int kNX  = 16;
constexpr int kNU  = 8;
constexpr int kNSC = 24;   // NX + NU
constexpr int kT   = 64;
constexpr int kNB  = 2048;

// One V_WMMA_F32_16X16X4_F32 step: D(16x16) = A(16x4) * B(4x16) + C
__device__ __forceinline__ v8f wmma_k4(v2f a, v2f b, v8f c) {
  return __builtin_amdgcn_wmma_f32_16x16x4_f32(
      /*neg_a=*/false, a, /*neg_b=*/false, b,
      /*c_mod=*/(short)0, c, /*reuse_a=*/false, /*reuse_b=*/false);
}

__global__ __launch_bounds__(32)
void mpc_lqr_kernel(const float* __restrict__ Qg,   // (NB, T, 24, 24)
                    const float* __restrict__ pg,   // (NB, T, 24)
                    const float* __restrict__ Ag,   // (NB, 16, 16)
                    const float* __restrict__ Bg,   // (NB, 16, 8)
                    const float* __restrict__ x0g,  // (NB, 16)
                    float* __restrict__ xtraj,      // (NB, T+1, 16)
                    float* __restrict__ usg,        // (NB, T, 8)
                    float* __restrict__ csg)        // (NB,)
{
  // Per-workgroup (one wave) LDS: ~44 KB
  __shared__ float Ks[kT][kNU][kNX];     // 32 KB gains
  __shared__ float ks[kT][kNU];          // 2 KB
  __shared__ float F_s[kNX][32];         // F=[A|B], cols 24..31 zero padded
  __shared__ float W_s[kNX][32];         // W = V*F
  __shared__ float V_s[kNX][kNX];
  __shared__ float Qt_s[kNSC][kNSC + 1]; // Qt = Q + F^T V F
  __shared__ float M_s[kNU][28];         // augmented [Quu | Qux | qu]
  __shared__ float qt_s[kNSC];
  __shared__ float v_s[kNX];
  __shared__ float x_s[kNX];
  __shared__ float xu_s[kNSC];

  const int lane = threadIdx.x;
  const int b    = blockIdx.x;
  const int hi   = (lane >= 16) ? 1 : 0;
  const int l15  = lane & 15;

  const float* Qb_base = Qg + (size_t)b * kT * kNSC * kNSC;
  const float* pb_base = pg + (size_t)b * kT * kNSC;

  // ---- Load F = [A | B] into LDS, zero-pad columns 24..31 ----
  for (int idx = lane; idx < kNX * 32; idx += 32) {
    int r = idx >> 5, c = idx & 31;
    float v = 0.0f;
    if (c < kNX)       v = Ag[(size_t)b * kNX * kNX + r * kNX + c];
    else if (c < kNSC) v = Bg[(size_t)b * kNX * kNU + r * kNU + (c - kNX)];
    F_s[r][c] = v;
  }
  for (int idx = lane; idx < kNX * kNX; idx += 32) V_s[idx >> 4][idx & 15] = 0.0f;
  if (lane < kNX) v_s[lane] = 0.0f;
  __syncthreads();

  // ================= Backward Riccati recursion =================
  for (int t = kT - 1; t >= 0; --t) {
    const float* Qt_g = Qb_base + (size_t)t * kNSC * kNSC;
    const float* pt_g = pb_base + (size_t)t * kNSC;

    // Prefetch next timestep's Q block (2304 B)
    if (t > 0) {
      const char* nxt = (const char*)(Qt_g - kNSC * kNSC);
      if (lane * 128 < kNSC * kNSC * 4)
        __builtin_prefetch(nxt + lane * 128, 0, 1);
    }

    // qt = p + F^T v  (uses previous v)
    if (lane < kNSC) {
      float s = pt_g[lane];
      for (int x = 0; x < kNX; ++x) s += F_s[x][lane] * v_s[x];
      qt_s[lane] = s;
    }

    // ---- W = V * F   (16 x 32, padded cols are zero) ----
    for (int ntile = 0; ntile < 2; ++ntile) {
      const int n0 = ntile * 16;
      v8f acc = {};
      for (int k0 = 0; k0 < 16; k0 += 4) {
        const int kA = k0 + (hi ? 2 : 0);
        v2f a, bb;
        a.x  = V_s[l15][kA];     a.y  = V_s[l15][kA + 1];
        bb.x = F_s[kA][n0 + l15]; bb.y = F_s[kA + 1][n0 + l15];
        acc = wmma_k4(a, bb, acc);
      }
      for (int r = 0; r < 8; ++r) W_s[r + (hi ? 8 : 0)][n0 + l15] = acc[r];
    }
    __syncthreads();

    // ---- Qt = Q[t] + F^T * W   (24x24, computed as 2x2 16-tiles) ----
    for (int mtile = 0; mtile < 2; ++mtile) {
      for (int ntile = 0; ntile < 2; ++ntile) {
        const int m0 = mtile * 16, n0 = ntile * 16;
        v8f acc;
        for (int r = 0; r < 8; ++r) {  // C initialized directly from global Q
          const int row = m0 + r + (hi ? 8 : 0);
          const int col = n0 + l15;
          acc[r] = (row < kNSC && col < kNSC) ? Qt_g[row * kNSC + col] : 0.0f;
        }
        for (int k0 = 0; k0 < 16; k0 += 4) {
          const int kA = k0 + (hi ? 2 : 0);
          v2f a, bb;
          a.x  = F_s[kA][m0 + l15]; a.y  = F_s[kA + 1][m0 + l15]; // (F^T)[m][k]
          bb.x = W_s[kA][n0 + l15]; bb.y = W_s[kA + 1][n0 + l15];
          acc = wmma_k4(a, bb, acc);
        }
        for (int r = 0; r < 8; ++r) {
          const int row = m0 + r + (hi ? 8 : 0);
          const int col = n0 + l15;
          if (row < kNSC && col < kNSC) Qt_s[row][col] = acc[r];
        }
      }
    }
    __syncthreads();

    // ---- Build augmented system [Quu | Qux | qu], 8 rows x 25 cols ----
    if (lane < 25) {
      for (int r = 0; r < kNU; ++r) {
        float vv;
        if (lane < 8)       vv = Qt_s[kNX + r][kNX + lane];
        else if (lane < 24) vv = Qt_s[kNX + r][lane - 8];
        else                vv = qt_s[kNX + r];
        M_s[r][lane] = vv;
      }
    }
    __syncthreads();

    // ---- Gauss-Jordan (Quu is SPD: no pivoting needed) ----
    for (int i = 0; i < kNU; ++i) {
      const float pivinv = 1.0f / M_s[i][i];
      if (lane < 25) M_s[i][lane] *= pivinv;
      __syncthreads();
      for (int r = 0; r < kNU; ++r) {
        if (r == i) continue;
        const float f = M_s[r][i];
        if (lane < 25) M_s[r][lane] -= f * M_s[i][lane];
      }
      __syncthreads();
    }

    // K = -Quu^{-1} Qux ; k = -Quu^{-1} qu
    if (lane < kNX)
      for (int r = 0; r < kNU; ++r) Ks[t][r][lane] = -M_s[r][8 + lane];
    if (lane < kNU) ks[t][lane] = -M_s[lane][24];
    __syncthreads();

    // ---- Vn = Qxx + Qxu*K  (exact since Quu*K = -Qux) ----
    {
      v8f acc;
      for (int r = 0; r < 8; ++r) acc[r] = Qt_s[r + (hi ? 8 : 0)][l15]; // Qxx
      for (int k0 = 0; k0 < 8; k0 += 4) {
        const int kA = k0 + (hi ? 2 : 0);
        v2f a, bb;
        a.x  = Qt_s[l15][kNX + kA]; a.y = Qt_s[l15][kNX + kA + 1]; // Qxu
        bb.x = Ks[t][kA][l15];      bb.y = Ks[t][kA + 1][l15];
        acc = wmma_k4(a, bb, acc);
      }
      for (int r = 0; r < 8; ++r) V_s[r + (hi ? 8 : 0)][l15] = acc[r];
    }
    // vn = qx + Qxu*k
    if (lane < kNX) {
      float s = qt_s[lane];
      for (int u = 0; u < kNU; ++u) s += Qt_s[lane][kNX + u] * ks[t][u];
      v_s[lane] = s;
    }
    __syncthreads();
  }

  // ================= Forward rollout =================
  if (lane < kNX) {
    const float xi = x0g[(size_t)b * kNX + lane];
    x_s[lane] = xi;
    xtraj[(size_t)b * (kT + 1) * kNX + lane] = xi;
  }
  float csum = 0.0f;
  __syncthreads();

  for (int t = 0; t < kT; ++t) {
    const float* Qt_g = Qb_base + (size_t)t * kNSC * kNSC;
    const float* pt_g = pb_base + (size_t)t * kNSC;

    // u = K x + k
    if (lane < kNU) {
      float u = ks[t][lane];
      for (int x = 0; x < kNX; ++x) u += Ks[t][lane][x] * x_s[x];
      xu_s[kNX + lane] = u;
      usg[(size_t)b * kT * kNU + (size_t)t * kNU + lane] = u;
    }
    if (lane < kNX) xu_s[lane] = x_s[lane];
    __syncthreads();

    // cost: c = 0.5 * xu^T Q xu + p . xu
    float ci = 0.0f;
    if (lane < kNSC) {
      float rd = 0.0f;
      for (int j = 0; j < kNSC; ++j) rd += Qt_g[lane * kNSC + j] * xu_s[j];
      ci = xu_s[lane] * (0.5f * rd + pt_g[lane]);
    }
    for (int off = 16; off > 0; off >>= 1) ci += __shfl_xor(ci, off, 32);
    csum += ci;

    // xn = F * xu
    float xn = 0.0f;
    if (lane < kNX)
      for (int j = 0; j < kNSC; ++j) xn += F_s[lane][j] * xu_s[j];
    __syncthreads();
    if (lane < kNX) {
      x_s[lane] = xn;
      xtraj[(size_t)b * (kT + 1) * kNX + (size_t)(t + 1) * kNX + lane] = xn;
    }
    __syncthreads();
  }
  if (lane == 0) csg[b] = csum;
}

extern "C" void kernel_launch(void* const* d_in, const int* in_sizes, int n_in,
                              void* d_out, int out_size, void* d_ws, size_t ws_size,
                              hipStream_t stream) {
  (void)in_sizes; (void)n_in; (void)out_size; (void)d_ws; (void)ws_size;
  const float* Q  = (const float*)d_in[0];
  const float* p  = (const float*)d_in[1];
  const float* A  = (const float*)d_in[2];
  const float* Bm = (const float*)d_in[3];
  const float* x0 = (const float*)d_in[4];

  float* out   = (float*)d_out;
  float* xtraj = out;                                           // NB*(T+1)*NX
  float* us    = xtraj + (size_t)kNB * (kT + 1) * kNX;          // NB*T*NU
  float* cs    = us + (size_t)kNB * kT * kNU;                   // NB

  hipLaunchKernelGGL(mpc_lqr_kernel, dim3(kNB), dim3(32), 0, stream,
                     Q, p, A, Bm, x0, xtraj, us, cs);
}